// RecurrentDynamical3D_35527969472537
// MI455X (gfx1250) — compile-verified
//
#include <hip/hip_runtime.h>
#include <math.h>
#include <stdint.h>

#define B_TOT 65536
#define T_LEN 128
#define BPB   64                 // batches (threads) per block
#define NBLK  (B_TOT / BPB)      // 1024 blocks
#define CT    16                 // timesteps per staged chunk
#define NC    (T_LEN / CT)       // 8 chunks
#define XROW  52                 // padded floats per batch per chunk (48 data + 4 pad)
#define HSTR  18                 // h-history row stride (bank-conflict-free, 8B aligned)

typedef float v2f __attribute__((ext_vector_type(2)));
typedef float v8f __attribute__((ext_vector_type(8)));

#if defined(__gfx1250__) && __has_builtin(__builtin_amdgcn_wmma_f32_16x16x4_f32)
#define USE_WMMA 1
#else
#define USE_WMMA 0
#endif

// ---- CDNA5 async global->LDS staging (ASYNCcnt path) ----
#if defined(__gfx1250__) && __has_builtin(__builtin_amdgcn_global_load_async_to_lds_b128)
#define HAS_ASYNC 1
// Builtin prototype (from compiler diagnostic): first param is
//   __attribute__((vector_size(16))) int  in address_space(1)  (global src)
// second is the LDS destination. Match the pointee types exactly.
typedef int v16b __attribute__((vector_size(16)));
typedef __attribute__((address_space(1))) v16b gv16b;
typedef __attribute__((address_space(3))) v16b lv16b;
#if __has_builtin(__builtin_amdgcn_s_wait_asynccnt)
#define WAIT_ASYNC(n) __builtin_amdgcn_s_wait_asynccnt(n)
#else
#define WAIT_ASYNC(n) asm volatile("s_wait_asynccnt " #n ::: "memory")
#endif
#else
#define HAS_ASYNC 0
#define WAIT_ASYNC(n)
#endif

struct __align__(16) SMem {
  float xbuf[2][BPB * XROW];     // staged input, double buffered (26 KB)
  float wbuf[T_LEN * 16];        // Wcat, k-major, N padded to 16 (8 KB)
#if USE_WMMA
  float hc[2][32 * HSTR];        // per-wave h history for current chunk
  float outb[BPB * 16];          // unloaded WMMA D
#endif
  float red[BPB];
};

// ---- fast nonlinearities: gfx1250 has V_TANH_F32 ----
#if defined(__gfx1250__) && __has_builtin(__builtin_amdgcn_tanhf)
__device__ __forceinline__ float fast_tanh(float x) { return __builtin_amdgcn_tanhf(x); }
#elif defined(__gfx1250__) && __has_builtin(__builtin_amdgcn_tanh_f32)
__device__ __forceinline__ float fast_tanh(float x) { return __builtin_amdgcn_tanh_f32(x); }
#else
__device__ __forceinline__ float fast_tanh(float x) {
  const float xc = fminf(fmaxf(x, -20.0f), 20.0f);
  const float e  = __expf(2.0f * xc);
  return (e - 1.0f) / (e + 1.0f);
}
#endif
__device__ __forceinline__ float fast_sigmoid(float x) {
  return 0.5f * fast_tanh(0.5f * x) + 0.5f;
}

// Stage one 16-timestep chunk for 64 batches into LDS (coalesced b128).
__device__ __forceinline__ void stage_chunk(const float* __restrict__ hist,
                                            float* __restrict__ dst,
                                            int b0, int cc, int tid) {
#pragma unroll
  for (int it = 0; it < 12; ++it) {
    const int q  = it * BPB + tid;
    const int bl = q / 12, j = q % 12;
    const float* g = hist + (size_t)(b0 + bl) * 384 + cc * 48 + j * 4;
    float*       l = dst + bl * XROW + j * 4;
#if HAS_ASYNC
    __builtin_amdgcn_global_load_async_to_lds_b128(
        (gv16b*)(size_t)g, (lv16b*)(unsigned)(size_t)l, 0, 0);
#else
    *(float4*)l = *(const float4*)g;
#endif
  }
}

__global__ __launch_bounds__(BPB) void rd3d_main(
    const float* __restrict__ hist, const float* __restrict__ h0v,
    const float* __restrict__ c0v,  const float* __restrict__ tgt,
    const float* __restrict__ Wih,  const float* __restrict__ Whh,
    const float* __restrict__ bih,  const float* __restrict__ bhh,
    const float* __restrict__ Wm,   const float* __restrict__ bm,
    const float* __restrict__ Wv,   const float* __restrict__ bv,
    float* __restrict__ partial)
{
  __shared__ SMem sm;
  const int tid = threadIdx.x;
  const int wv  = tid >> 5;
  const int ln  = tid & 31;
  const int b0  = blockIdx.x * BPB;
  const int b   = b0 + tid;

  // ---- issue async staging for chunks 0 and 1 first (overlaps all setup) ----
  stage_chunk(hist, sm.xbuf[0], b0, 0, tid);
  stage_chunk(hist, sm.xbuf[1], b0, 1, tid);

  // ---- Wcat -> LDS (rows 0-2 = W_mean, 3-8 = W_var, 9-15 = 0), k-major ----
  for (int i = tid; i < T_LEN * 16; i += BPB) {
    const int k = i >> 4, n = i & 15;
    float w = 0.0f;
    if (n < 3)      w = Wm[n * T_LEN + k];
    else if (n < 9) w = Wv[(n - 3) * T_LEN + k];
    sm.wbuf[i] = w;
  }

  // ---- uniform LSTM weights (scalar loads) ----
  float wih[12], whh[4], bb[4];
#pragma unroll
  for (int q = 0; q < 4; ++q) {
    wih[q * 3 + 0] = Wih[q * 3 + 0];
    wih[q * 3 + 1] = Wih[q * 3 + 1];
    wih[q * 3 + 2] = Wih[q * 3 + 2];
    whh[q] = Whh[q];
    bb[q]  = bih[q] + bhh[q];
  }

  // ---- rotation matrix from rows t = 127, 126, 125 ----
  const float* xb = hist + (size_t)b * (T_LEN * 3);
  const float v1x = xb[381], v1y = xb[382], v1z = xb[383];
  const float v2x = xb[378], v2y = xb[379], v2z = xb[380];
  const float v3x = xb[375], v3y = xb[376], v3z = xb[377];
  const float i1  = 1.0f / sqrtf(v1x * v1x + v1y * v1y + v1z * v1z);
  const float b1x = v1x * i1, b1y = v1y * i1, b1z = v1z * i1;
  const float dp  = v2x * b1x + v2y * b1y + v2z * b1z;
  const float a2x = v2x - dp * b1x, a2y = v2y - dp * b1y, a2z = v2z - dp * b1z;
  const float i2  = 1.0f / sqrtf(a2x * a2x + a2y * a2y + a2z * a2z);
  const float b2x = a2x * i2, b2y = a2y * i2, b2z = a2z * i2;
  float b3x = b1y * b2z - b1z * b2y;
  float b3y = b1z * b2x - b1x * b2z;
  float b3z = b1x * b2y - b1y * b2x;
  const float sgn = (v3x * b3x + v3y * b3y + v3z * b3z) > 0.0f ? 1.0f : -1.0f;
  b3x *= sgn; b3y *= sgn; b3z *= sgn;

  const float t0 = tgt[b * 3 + 0], t1 = tgt[b * 3 + 1], t2 = tgt[b * 3 + 2];
  const float rt0 = t0 * b1x + t1 * b1y + t2 * b1z;
  const float rt1 = t0 * b2x + t1 * b2y + t2 * b2z;
  const float rt2 = t0 * b3x + t1 * b3y + t2 * b3z;

  float h = h0v[b], c = c0v[b];

#if USE_WMMA
  v8f acc0 = {0, 0, 0, 0, 0, 0, 0, 0};   // batches [wave*32, wave*32+16)
  v8f acc1 = {0, 0, 0, 0, 0, 0, 0, 0};   // batches [wave*32+16, wave*32+32)
#else
  float proj[9];
#pragma unroll
  for (int j = 0; j < 9; ++j) proj[j] = 0.0f;
#endif

  for (int cc = 0; cc < NC; ++cc) {
    const int buf = cc & 1;
    // chunk cc complete for this wave (async completes in order); newer chunk
    // (12 per-wave async instructions) may remain outstanding.
    if (cc + 1 < NC) { WAIT_ASYNC(12); }
    else             { WAIT_ASYNC(0);  }
    __syncthreads();                       // chunk cc visible block-wide

    const float* xr = &sm.xbuf[buf][tid * XROW];
#pragma unroll
    for (int s = 0; s < CT; ++s) {
      const float x0 = xr[s * 3 + 0], x1 = xr[s * 3 + 1], x2 = xr[s * 3 + 2];
      const float rx = x0 * b1x + x1 * b1y + x2 * b1z;
      const float ry = x0 * b2x + x1 * b2y + x2 * b2z;
      const float rz = x0 * b3x + x1 * b3y + x2 * b3z;
      const float g0 = wih[0] * rx + wih[1]  * ry + wih[2]  * rz + whh[0] * h + bb[0];
      const float g1 = wih[3] * rx + wih[4]  * ry + wih[5]  * rz + whh[1] * h + bb[1];
      const float g2 = wih[6] * rx + wih[7]  * ry + wih[8]  * rz + whh[2] * h + bb[2];
      const float g3 = wih[9] * rx + wih[10] * ry + wih[11] * rz + whh[3] * h + bb[3];
      const float ig = fast_sigmoid(g0);
      const float fg = fast_sigmoid(g1);
      const float gg = fast_tanh(g2);
      const float og = fast_sigmoid(g3);
      c = fg * c + ig * gg;
      h = og * fast_tanh(c);
#if USE_WMMA
      sm.hc[wv][ln * HSTR + s] = h;        // bank-conflict-free strip
#else
      const int t = cc * CT + s;
#pragma unroll
      for (int j = 0; j < 9; ++j) proj[j] += h * sm.wbuf[t * 16 + j];
#endif
    }

#if USE_WMMA
    // feat @ Wcat^T via V_WMMA_F32_16X16X4_F32, K consumed 4 at a time.
    // A 16x4: lane m=l%16, VGPR0/1 = K pair selected by l/16.
    // B 4x16: lane n=l%16, VGPR0/1 = same K pair.
    {
      const int n  = ln & 15;
      const int kh = ln >> 4;
#pragma unroll
      for (int kk = 0; kk < 4; ++kk) {
        const int kloc = kk * 4 + 2 * kh;
        const int kabs = cc * CT + kloc;
        v2f bvec;
        bvec.x = sm.wbuf[kabs * 16 + n];
        bvec.y = sm.wbuf[(kabs + 1) * 16 + n];
        v2f a0;
        a0.x = sm.hc[wv][n * HSTR + kloc];
        a0.y = sm.hc[wv][n * HSTR + kloc + 1];
        acc0 = __builtin_amdgcn_wmma_f32_16x16x4_f32(false, a0, false, bvec,
                                                     (short)0, acc0, false, false);
        v2f a1;
        a1.x = sm.hc[wv][(16 + n) * HSTR + kloc];
        a1.y = sm.hc[wv][(16 + n) * HSTR + kloc + 1];
        acc1 = __builtin_amdgcn_wmma_f32_16x16x4_f32(false, a1, false, bvec,
                                                     (short)0, acc1, false, false);
      }
    }
#endif
    __syncthreads();                       // all threads done reading buf
    if (cc + 2 < NC) stage_chunk(hist, sm.xbuf[buf], b0, cc + 2, tid);
  }

#if USE_WMMA
  {
    // D layout: VGPR r holds M = r + 8*(lane/16), N = lane%16
    const int n  = ln & 15;
    const int mo = (ln >> 4) << 3;
#pragma unroll
    for (int r = 0; r < 8; ++r) {
      sm.outb[(wv * 32 +      r + mo) * 16 + n] = acc0[r];
      sm.outb[(wv * 32 + 16 + r + mo) * 16 + n] = acc1[r];
    }
  }
  __syncthreads();
  float proj[9];
#pragma unroll
  for (int j = 0; j < 9; ++j) proj[j] = sm.outb[tid * 16 + j];
#endif

  // ---- NLL tail: Sigma = L D L^T  ->  quad via forward substitution ----
  const float m0 = proj[0] + bm[0];
  const float m1 = proj[1] + bm[1];
  const float m2 = proj[2] + bm[2];
  const float e0 = proj[3] + bv[0];
  const float e1 = proj[4] + bv[1];
  const float e2 = proj[5] + bv[2];
  const float dv0 = __expf(proj[6] + bv[3]);
  const float dv1 = __expf(proj[7] + bv[4]);
  const float dv2 = __expf(proj[8] + bv[5]);
  const float d0 = m0 - rt0, d1 = m1 - rt1, d2 = m2 - rt2;
  const float y0 = d0;
  const float y1 = d1 - e0 * y0;
  const float y2 = d2 - e1 * y0 - e2 * y1;
  const float quad = y0 * y0 / dv0 + y1 * y1 / dv1 + y2 * y2 / dv2;
  const float loc  = 0.5f * (dv0 + dv1 + dv2 + quad);

  // ---- deterministic block reduction ----
  sm.red[tid] = loc;
  __syncthreads();
#pragma unroll
  for (int off = BPB / 2; off > 0; off >>= 1) {
    if (tid < off) sm.red[tid] += sm.red[tid + off];
    __syncthreads();
  }
  if (tid == 0) partial[blockIdx.x] = sm.red[0];
}

__global__ __launch_bounds__(256) void rd3d_reduce(const float* __restrict__ partial,
                                                   float* __restrict__ out)
{
  __shared__ float r[256];
  float s = 0.0f;
  for (int i = threadIdx.x; i < NBLK; i += 256) s += partial[i];
  r[threadIdx.x] = s;
  __syncthreads();
#pragma unroll
  for (int off = 128; off > 0; off >>= 1) {
    if (threadIdx.x < off) r[threadIdx.x] += r[threadIdx.x + off];
    __syncthreads();
  }
  if (threadIdx.x == 0) out[0] = r[0] / (float)B_TOT;
}

extern "C" void kernel_launch(void* const* d_in, const int* in_sizes, int n_in,
                              void* d_out, int out_size, void* d_ws, size_t ws_size,
                              hipStream_t stream) {
  const float* hist = (const float*)d_in[0];
  const float* h0v  = (const float*)d_in[1];
  const float* c0v  = (const float*)d_in[2];
  const float* tgt  = (const float*)d_in[3];
  const float* Wih  = (const float*)d_in[4];
  const float* Whh  = (const float*)d_in[5];
  const float* bih  = (const float*)d_in[6];
  const float* bhh  = (const float*)d_in[7];
  const float* Wm   = (const float*)d_in[8];
  const float* bm   = (const float*)d_in[9];
  const float* Wv   = (const float*)d_in[10];
  const float* bv   = (const float*)d_in[11];
  float* out     = (float*)d_out;
  float* partial = (float*)d_ws;

  rd3d_main<<<NBLK, BPB, 0, stream>>>(hist, h0v, c0v, tgt, Wih, Whh, bih, bhh,
                                      Wm, bm, Wv, bv, partial);
  rd3d_reduce<<<1, 256, 0, stream>>>(partial, out);
}